// EfficientInteractionBilinear_76639396429997
// MI455X (gfx1250) — compile-verified
//
#include <hip/hip_runtime.h>

// ---------------------------------------------------------------------------
// EfficientInteractionBilinear for MI455X (gfx1250, wave32, f32 WMMA 16x16x4)
//
//   m_pad  = scatter_add(m at (id_reduce, Kidx))        (E,24,64)   [atomics]
//   sum_k  = sph^T @ m_pad        per edge  (16x24)(24x64) -> (16,64)
//   X      = rbf_W1 @ sum_k       per edge  (64x16)(16x64) -> (64,64)
//   out    = contract(X, W2)      16-edge tile GEMM (16x4096)(4096x64)
// ---------------------------------------------------------------------------

typedef __attribute__((ext_vector_type(2))) float v2f;
typedef __attribute__((ext_vector_type(8))) float v8f;

#define E_TOT 100000
#define KMAX  24
#define T_TOT 1200000
#define NSPH  16
#define EMB   64     // h
#define INTM  64     // i
#define OUT   64     // o
#define TE    16     // edges per workgroup

// LDS layout (floats), dynamic shared memory:
//   sum_k : [TE][16 s][65]          (row stride 65 to de-conflict banks)
#define SK_ESTRIDE  (NSPH * (EMB + 1))        // 1040
#define SK_SIZE     (TE * SK_ESTRIDE)         // 16640
//   A3    : [TE][1028]  X-tile as stage-3 A matrix, c = i*16 + h_local
#define A3_ESTRIDE  1028
#define A3_OFF      SK_SIZE
#define A3_SIZE     (TE * A3_ESTRIDE)         // 16448
//   red   : [8 waves][16 e][16 o_local]
#define RED_OFF     (A3_OFF + A3_SIZE)
#define RED_SIZE    (8 * TE * 16)             // 2048
#define LDS_FLOATS  (RED_OFF + RED_SIZE)      // 35136 floats = 140544 B

__device__ __forceinline__ v8f wmma_f32(v2f a, v2f b, v8f c) {
  // D(16x16,f32) = A(16x4,f32) x B(4x16,f32) + C
  return __builtin_amdgcn_wmma_f32_16x16x4_f32(
      /*neg_a=*/false, a, /*neg_b=*/false, b,
      /*c_mod=*/(short)0, c, /*reuse_a=*/false, /*reuse_b=*/false);
}

// ---------------------------------------------------------------------------
// Kernel 1: scatter-add m -> m_pad (duplicates accumulate, tf.scatter_nd)
// ---------------------------------------------------------------------------
__global__ __launch_bounds__(256) void scatter_kernel(
    const float* __restrict__ m, const int* __restrict__ id_reduce,
    const int* __restrict__ Kidx, float* __restrict__ m_pad) {
  unsigned idx = blockIdx.x * 256u + threadIdx.x;   // T_TOT*16 threads
  unsigned t = idx >> 4;
  unsigned h = (idx & 15u) << 2;                    // 4 floats per thread
  if (t >= T_TOT) return;
  const float4 val = *reinterpret_cast<const float4*>(m + (size_t)t * EMB + h);
  float* dst = m_pad + ((size_t)id_reduce[t] * KMAX + (size_t)Kidx[t]) * EMB + h;
  atomicAdd(dst + 0, val.x);
  atomicAdd(dst + 1, val.y);
  atomicAdd(dst + 2, val.z);
  atomicAdd(dst + 3, val.w);
}

// ---------------------------------------------------------------------------
// Kernel 2: fused per-16-edge pipeline, all stages in f32 WMMA
// ---------------------------------------------------------------------------
__global__ __launch_bounds__(256) void fused_kernel(
    const float* __restrict__ rbf_W1,   // [E][64 i][16 s]
    const float* __restrict__ sph,      // [E][24 k][16 s]
    const float* __restrict__ W2,       // [64 h][64 i][64 o]
    const float* __restrict__ m_pad,    // [E][24 k][64 h]
    float* __restrict__ m_ca)           // [E][64 o]
{
  extern __shared__ float lds[];
  float* lds_sumk = lds;            // SK_SIZE
  float* lds_A    = lds + A3_OFF;   // A3_SIZE
  float* lds_red  = lds + RED_OFF;  // RED_SIZE

  const int tid  = threadIdx.x;
  const int wave = tid >> 5;
  const int lane = tid & 31;
  const int lh   = lane & 15;
  const int hi   = (lane >> 4) & 1;       // upper half-wave
  const size_t e0 = (size_t)blockIdx.x * TE;

  // ------------------------------------------------------------------
  // Stage 1: sum_k[e] = sph[e]^T (16x24) @ m_pad[e] (24x64)  -> LDS
  // wave w handles edges {2w, 2w+1}
  // ------------------------------------------------------------------
  for (int ee = 0; ee < 2; ++ee) {
    const int e_loc = wave * 2 + ee;
    const size_t e = e0 + e_loc;
    const float* sph_e = sph   + e * (size_t)(KMAX * NSPH);
    const float* mp_e  = m_pad + e * (size_t)(KMAX * EMB);
    float* dst = lds_sumk + e_loc * SK_ESTRIDE;
#pragma unroll
    for (int nt = 0; nt < 4; ++nt) {
      v8f acc = {};
#pragma unroll
      for (int k0 = 0; k0 < KMAX; k0 += 4) {
        const int ka = k0 + hi * 2;
        v2f a;  // A[s][k] = sph_e[k][s] : lanes = s, halves split K
        a.x = sph_e[ka * NSPH + lh];
        a.y = sph_e[(ka + 1) * NSPH + lh];
        v2f b;  // B[k][n] = mp_e[k][nt*16+n] : lanes = n, halves split K
        b.x = mp_e[ka * EMB + nt * 16 + lh];
        b.y = mp_e[(ka + 1) * EMB + nt * 16 + lh];
        acc = wmma_f32(a, b, acc);
      }
#pragma unroll
      for (int r = 0; r < 8; ++r) {   // D rows: r (lo half) / r+8 (hi half)
        const int s = hi ? r + 8 : r;
        dst[s * (EMB + 1) + nt * 16 + lh] = acc[r];
      }
    }
  }

  // Preload rbf_W1 A-fragments for stage 2 (reused across all 4 h-tiles)
  v2f rbfA[2][4][4];  // [edge][i-tile][s-step]
#pragma unroll
  for (int ee = 0; ee < 2; ++ee) {
    const size_t e = e0 + wave * 2 + ee;
    const float* rbf_e = rbf_W1 + e * (size_t)(INTM * NSPH);
#pragma unroll
    for (int mt = 0; mt < 4; ++mt) {
#pragma unroll
      for (int ks = 0; ks < 4; ++ks) {
        const int i  = mt * 16 + lh;
        const int s0 = ks * 4 + hi * 2;
        rbfA[ee][mt][ks].x = rbf_e[i * NSPH + s0];
        rbfA[ee][mt][ks].y = rbf_e[i * NSPH + s0 + 1];
      }
    }
  }

  __syncthreads();

  // ------------------------------------------------------------------
  // Stages 2+3, streamed over 4 h-tiles of 16:
  //   X_tile[e](64i x 16h) = rbf_W1[e] @ sum_k[e][:, ht]      (per wave: 2 edges)
  //   out(16e x 64o) += X(16e x 1024c) @ W2[(ht*16+hl)*64+i][o]  (c = i*16+hl)
  // wave w owns o-tile (w&3) and K-half (w>>2) of stage 3.
  // ------------------------------------------------------------------
  const int nt3 = wave & 3;
  const int kh  = wave >> 2;
  v8f acc3 = {};

  for (int ht = 0; ht < 4; ++ht) {
    // ---- Stage 2: write X tile into lds_A as stage-3 A layout ----
#pragma unroll
    for (int ee = 0; ee < 2; ++ee) {
      const int e_loc = wave * 2 + ee;
      const float* sk = lds_sumk + e_loc * SK_ESTRIDE;
      float* Adst = lds_A + e_loc * A3_ESTRIDE;
#pragma unroll
      for (int mt = 0; mt < 4; ++mt) {
        v8f acc = {};
#pragma unroll
        for (int ks = 0; ks < 4; ++ks) {
          const int k = ks * 4 + hi * 2;
          v2f b;  // B[s][n] = sum_k[s][ht*16+n]
          b.x = sk[k * (EMB + 1) + ht * 16 + lh];
          b.y = sk[(k + 1) * (EMB + 1) + ht * 16 + lh];
          acc = wmma_f32(rbfA[ee][mt][ks], b, acc);
        }
#pragma unroll
        for (int r = 0; r < 8; ++r) {
          const int i = mt * 16 + (hi ? r + 8 : r);
          Adst[i * 16 + lh] = acc[r];   // c = i*16 + h_local
        }
      }
    }
    __syncthreads();

    // ---- Stage 3: out += X_chunk @ W2_chunk ----
    const float* W2ht = W2 + (size_t)ht * 16 * INTM * OUT;
    for (int kc = 0; kc < 128; ++kc) {
      const int c = kh * 512 + kc * 4 + hi * 2;   // K index (i*16 + hl)
      const int i  = c >> 4;
      const int hl = c & 15;
      v2f a;  // A[e][c]: lanes = e (edge), halves split K
      a.x = lds_A[lh * A3_ESTRIDE + c];
      a.y = lds_A[lh * A3_ESTRIDE + c + 1];
      v2f b;  // B[c][o] = W2[ht*16+hl][i][nt3*16+o]
      b.x = W2ht[((size_t)hl * INTM + i) * OUT + nt3 * 16 + lh];
      b.y = W2ht[((size_t)(hl + 1) * INTM + i) * OUT + nt3 * 16 + lh];
      acc3 = wmma_f32(a, b, acc3);
    }
    __syncthreads();  // lds_A will be overwritten next h-tile
  }

  // ------------------------------------------------------------------
  // Reduce the two K-half partials per o-tile and store m_ca
  // ------------------------------------------------------------------
  float* red = lds_red + wave * (TE * 16);
#pragma unroll
  for (int r = 0; r < 8; ++r) {
    const int e_ = hi ? r + 8 : r;
    red[e_ * 16 + lh] = acc3[r];
  }
  __syncthreads();

#pragma unroll
  for (int q = 0; q < 4; ++q) {
    const int idx = q * 256 + tid;        // 0..1023 over (e, o)
    const int e_ = idx >> 6;
    const int o  = idx & 63;
    const int nt = o >> 4;
    const int ol = o & 15;
    const float v = lds_red[nt * (TE * 16) + e_ * 16 + ol] +
                    lds_red[(nt + 4) * (TE * 16) + e_ * 16 + ol];
    m_ca[(e0 + e_) * OUT + o] = v;
  }
}

// ---------------------------------------------------------------------------
extern "C" void kernel_launch(void* const* d_in, const int* in_sizes, int n_in,
                              void* d_out, int out_size, void* d_ws, size_t ws_size,
                              hipStream_t stream) {
  const float* rbf_W1    = (const float*)d_in[0];
  const float* sph       = (const float*)d_in[1];
  const float* m         = (const float*)d_in[2];
  const float* W2        = (const float*)d_in[3];
  const int*   id_reduce = (const int*)d_in[4];
  const int*   Kidx      = (const int*)d_in[5];
  // d_in[6] = nEdges scalar (compile-time constant E_TOT here)

  float* m_pad = (float*)d_ws;  // E_TOT*24*64 f32 = 614.4 MB scratch
  const size_t mpad_bytes = (size_t)E_TOT * KMAX * EMB * sizeof(float);
  hipMemsetAsync(m_pad, 0, mpad_bytes, stream);

  const unsigned scat_blocks = (unsigned)(((size_t)T_TOT * 16 + 255) / 256);
  scatter_kernel<<<scat_blocks, 256, 0, stream>>>(m, id_reduce, Kidx, m_pad);

  fused_kernel<<<E_TOT / TE, 256, LDS_FLOATS * sizeof(float), stream>>>(
      rbf_W1, sph, W2, m_pad, (float*)d_out);
}